// PixelBitFiLayer_28192165331402
// MI455X (gfx1250) — compile-verified
//
#include <hip/hip_runtime.h>

// Problem constants (match reference)
#define BATCH      128
#define IMG_H      512
#define IMG_W      512
#define IMG_C      3
#define NUM_FAULTS 64

// 512*512*3 = 786432 int32 per image; total 100,663,296 int32
static constexpr long long IMG_ELEMS = (long long)IMG_H * IMG_W * IMG_C;
static constexpr long long TOT_ELEMS = (long long)BATCH * IMG_ELEMS;
static constexpr long long TOT_VEC4  = TOT_ELEMS / 4;      // 25,165,824 int4 chunks

// Launch geometry for the streaming copy: exact tiling, no remainder.
// 25,165,824 = 24,576 blocks * 256 threads * 4 int4 per thread.
#define COPY_THREADS 256
#define COPY_UNROLL  4
static constexpr int COPY_BLOCKS =
    (int)(TOT_VEC4 / ((long long)COPY_THREADS * COPY_UNROLL));  // 24,576
static_assert((long long)COPY_BLOCKS * COPY_THREADS * COPY_UNROLL == TOT_VEC4,
              "copy tiling must be exact");

typedef int v4i __attribute__((ext_vector_type(4)));

// Kernel 1: straight-line streaming copy. Each thread moves 4 independent
// b128 chunks (64 B), block-contiguous for perfect coalescing. Four loads
// are issued back-to-back (clauseable) before any store waits, giving 4x
// memory-level parallelism per wave vs. the loop version. NT hints bypass
// the 192 MB L2 (stream is 805 MB, never re-read).
__global__ __launch_bounds__(COPY_THREADS) void pixelfi_stream_copy(
    const v4i* __restrict__ in, v4i* __restrict__ out) {
  long long base = (long long)blockIdx.x * (COPY_THREADS * COPY_UNROLL)
                 + threadIdx.x;
  v4i v0 = __builtin_nontemporal_load(&in[base + 0 * COPY_THREADS]);
  v4i v1 = __builtin_nontemporal_load(&in[base + 1 * COPY_THREADS]);
  v4i v2 = __builtin_nontemporal_load(&in[base + 2 * COPY_THREADS]);
  v4i v3 = __builtin_nontemporal_load(&in[base + 3 * COPY_THREADS]);
  __builtin_nontemporal_store(v0, &out[base + 0 * COPY_THREADS]);
  __builtin_nontemporal_store(v1, &out[base + 1 * COPY_THREADS]);
  __builtin_nontemporal_store(v2, &out[base + 2 * COPY_THREADS]);
  __builtin_nontemporal_store(v3, &out[base + 3 * COPY_THREADS]);
}

// Kernel 2: apply all 64 faults to all 128 images directly on the output.
// XOR is associative/commutative, so atomicXor reproduces the reference's
// sequential-scan semantics exactly, even when faults collide on a pixel.
// 8192 lanes -> GLOBAL_ATOMIC_XOR_B32 (non-returning); negligible cost.
__global__ __launch_bounds__(256) void pixelfi_apply_faults(
    const int* __restrict__ faults, int* __restrict__ out) {
  int t = blockIdx.x * blockDim.x + threadIdx.x;   // 0 .. BATCH*NUM_FAULTS-1
  if (t >= BATCH * NUM_FAULTS) return;
  int f = t & (NUM_FAULTS - 1);
  int b = t >> 6;
  int fh = faults[f * 4 + 0];
  int fw = faults[f * 4 + 1];
  int fc = faults[f * 4 + 2];
  int fb = faults[f * 4 + 3];
  long long off = (long long)b * IMG_ELEMS
                + (long long)fh * (IMG_W * IMG_C)
                + (long long)fw * IMG_C + fc;
  atomicXor(&out[off], 1 << fb);
}

extern "C" void kernel_launch(void* const* d_in, const int* in_sizes, int n_in,
                              void* d_out, int out_size, void* d_ws, size_t ws_size,
                              hipStream_t stream) {
  (void)in_sizes; (void)n_in; (void)out_size; (void)d_ws; (void)ws_size;

  const v4i* images = (const v4i*)d_in[0];   // int32 [B,H,W,C], 256B-aligned (hipMalloc)
  const int* faults = (const int*)d_in[1];   // int32 [64,4] = (h,w,c,bit)
  int*       out    = (int*)d_out;           // int32 output, written bit-exact

  // Streaming copy: 24,576 blocks x 8 wave32, 64 B per thread, exact cover.
  pixelfi_stream_copy<<<COPY_BLOCKS, COPY_THREADS, 0, stream>>>(
      images, (v4i*)out);

  // Fault scatter: 8192 atomic XORs; stream order guarantees it runs after
  // the copy has fully written the output.
  const int total_faults = BATCH * NUM_FAULTS;                // 8192
  pixelfi_apply_faults<<<(total_faults + 255) / 256, 256, 0, stream>>>(
      faults, out);
}